// MaskRCNN_13082470383794
// MI455X (gfx1250) — compile-verified
//
#include <hip/hip_runtime.h>
#include <hip/hip_bf16.h>
#include <math.h>

typedef __attribute__((ext_vector_type(2))) float v2f;
typedef __attribute__((ext_vector_type(8))) float v8f;

// ---------------------------------------------------------------------------
// WMMA f32 16x16x4 wrapper  (D = A(16x4) x B(4x16) + C)
// ---------------------------------------------------------------------------
__device__ __forceinline__ v8f wmma_f32(v2f a, v2f b, v8f c) {
    return __builtin_amdgcn_wmma_f32_16x16x4_f32(
        /*neg_a=*/false, a, /*neg_b=*/false, b,
        /*c_mod=*/(short)0, c, /*reuse_a=*/false, /*reuse_b=*/false);
}

// ---------------------------------------------------------------------------
// Generic WMMA GEMM, 2-deep ping-pong pipeline, optional K-split (blockIdx.z).
// One wave per block; wave owns MT m-tiles (16 rows) x 32 cols.
// Grid: (ceil(M/(16*MT)), N/32, ksplit).  kchunk must be a multiple of 8.
//   A[m][k] = A[m*am + k]       (k contiguous -> 8B loads, masked by mvf)
//   B[k][n] = B[k*bk + n*bn]    (bn==1 -> coalesced across lanes)
// ksplit==1: C[m*N+n] = relu(acc + bias[n]);  ksplit>1: raw partial at z*M*N.
// ---------------------------------------------------------------------------
template <int MT>
__global__ __launch_bounds__(32)
void gemm_wmma(const float* __restrict__ A, int am,
               const float* __restrict__ B, int bk, int bn,
               const float* __restrict__ bias,
               float* __restrict__ C,
               int M, int N, int K, int relu, int kchunk) {
    const int lane  = threadIdx.x & 31;
    const int row16 = lane & 15;
    const int half  = lane >> 4;
    const int koff  = half * 2;
    const int m_base = blockIdx.x * (MT * 16);
    const int n0     = blockIdx.y * 32;

    const int kbegin = blockIdx.z * kchunk;
    const int kend   = (kbegin + kchunk < K) ? (kbegin + kchunk) : K;
    const bool finalout = (gridDim.z == 1);
    float* Cout = C + (size_t)blockIdx.z * (size_t)M * N;

    v8f acc[MT][2];
    const v8f vzero = {0.f,0.f,0.f,0.f,0.f,0.f,0.f,0.f};
#pragma unroll
    for (int t = 0; t < MT; ++t) { acc[t][0] = vzero; acc[t][1] = vzero; }

    const size_t nidx[2] = { (size_t)(n0 + row16), (size_t)(n0 + 16 + row16) };
    float mvf[MT]; size_t arow[MT];
#pragma unroll
    for (int t = 0; t < MT; ++t) {
        const int m = m_base + t * 16 + row16;
        mvf[t]  = (m < M) ? 1.f : 0.f;           // mask instead of exec branch
        arow[t] = (size_t)((m < M) ? m : 0) * am; // clamped -> always safe
    }

    auto loadB = [&](v2f bf[2], int k0) {
        const int kk = k0 + koff;
#pragma unroll
        for (int nf = 0; nf < 2; ++nf) {
            bf[nf][0] = B[(size_t)kk       * bk + nidx[nf] * bn];
            bf[nf][1] = B[(size_t)(kk + 1) * bk + nidx[nf] * bn];
        }
    };
    auto loadA = [&](v2f af[MT], int k0) {
        const int kk = k0 + koff;
#pragma unroll
        for (int t = 0; t < MT; ++t) {
            v2f v = *(const v2f*)(A + arow[t] + kk);
            af[t] = v * mvf[t];                   // 0/1 mask, no exec juggling
        }
    };
    auto domm = [&](v2f af[MT], v2f bf[2]) {
#pragma unroll
        for (int t = 0; t < MT; ++t) {
            acc[t][0] = wmma_f32(af[t], bf[0], acc[t][0]);
            acc[t][1] = wmma_f32(af[t], bf[1], acc[t][1]);
        }
    };

    v2f b0[2], a0[MT], b1[2], a1[MT];
    loadB(b0, kbegin); loadA(a0, kbegin);
    int k0 = kbegin;
    for (; k0 + 8 < kend; k0 += 8) {
        loadB(b1, k0 + 4); loadA(a1, k0 + 4);
        domm(a0, b0);
        loadB(b0, k0 + 8); loadA(a0, k0 + 8);
        domm(a1, b1);
    }
    // epilogue: exactly two k-steps remain (kchunk % 8 == 0)
    loadB(b1, k0 + 4); loadA(a1, k0 + 4);
    domm(a0, b0);
    domm(a1, b1);

#pragma unroll
    for (int t = 0; t < MT; ++t)
#pragma unroll
        for (int nf = 0; nf < 2; ++nf)
#pragma unroll
            for (int r = 0; r < 8; ++r) {
                const int m = m_base + t * 16 + r + half * 8;
                const int n = n0 + nf * 16 + row16;
                if (m < M) {
                    float v = acc[t][nf][r];
                    if (finalout) {
                        if (bias) v += bias[n];
                        if (relu) v = fmaxf(v, 0.f);
                    }
                    Cout[(size_t)m * N + n] = v;
                }
            }
}

// ---------------------------------------------------------------------------
// Sum K-split partials + bias + relu.
// ---------------------------------------------------------------------------
__global__ void reduce_partials(const float* __restrict__ P, int S,
                                const float* __restrict__ bias,
                                float* __restrict__ out, int MN, int N, int relu) {
    const int id = blockIdx.x * blockDim.x + threadIdx.x;
    if (id >= MN) return;
    float s = 0.f;
    for (int i = 0; i < S; ++i) s += P[(size_t)i * MN + id];
    s += bias[id % N];
    if (relu) s = fmaxf(s, 0.f);
    out[id] = s;
}

// ---------------------------------------------------------------------------
// Implicit-im2col 3x3 conv (pad 1) as WMMA GEMM, t-major K order
// (k = t*Cin + c).  Tap loop hoists (dy,dx)/bounds/base; the hot c-loop is
// unconditional strided loads * 0/1 mask + WMMA, 2-deep ping-pong pipeline.
// Cin must be a multiple of 8.
// ---------------------------------------------------------------------------
template <int MT>
__global__ __launch_bounds__(32)
void conv3x3_gemm_wmma(const float* __restrict__ feat, int Bn, int Cin, int H, int W,
                       const float* __restrict__ wOIHW, const float* __restrict__ bias,
                       float* __restrict__ out, int N, int relu) {
    const int HW = H * W;
    const int M  = Bn * HW;
    const int K9 = Cin * 9;
    const int lane  = threadIdx.x & 31;
    const int row16 = lane & 15;
    const int half  = lane >> 4;
    const int koff  = half * 2;
    const int m_base = blockIdx.x * (MT * 16);
    const int n0     = blockIdx.y * 32;

    v8f acc[MT][2];
    const v8f vzero = {0.f,0.f,0.f,0.f,0.f,0.f,0.f,0.f};
#pragma unroll
    for (int t = 0; t < MT; ++t) { acc[t][0] = vzero; acc[t][1] = vzero; }

    int bb[MT], yy[MT], xx[MT]; bool mv[MT];
#pragma unroll
    for (int q = 0; q < MT; ++q) {
        int m = m_base + q * 16 + row16;
        mv[q] = (m < M);
        if (!mv[q]) m = 0;
        bb[q] = m / HW;
        const int p = m - bb[q] * HW;
        yy[q] = p / W;
        xx[q] = p - yy[q] * W;
    }
    const size_t nidx[2] = { (size_t)(n0 + row16), (size_t)(n0 + 16 + row16) };

    for (int t = 0; t < 9; ++t) {
        const int dy = t / 3 - 1;
        const int dx = t - (t / 3) * 3 - 1;

        float avf[MT];
        const float* ap[MT];
#pragma unroll
        for (int q = 0; q < MT; ++q) {
            const int vy = yy[q] + dy, vx = xx[q] + dx;
            const bool ok = mv[q] && vy >= 0 && vy < H && vx >= 0 && vx < W;
            avf[q] = ok ? 1.f : 0.f;
            const int cy = vy < 0 ? 0 : (vy >= H ? H - 1 : vy);   // clamped: safe
            const int cx = vx < 0 ? 0 : (vx >= W ? W - 1 : vx);
            ap[q] = feat + (size_t)bb[q] * Cin * HW + (size_t)cy * W + cx;
        }
        const float* w0 = wOIHW + nidx[0] * K9 + t;
        const float* w1 = wOIHW + nidx[1] * K9 + t;

        auto loadBc = [&](v2f bf[2], int c0) {
            const int c = c0 + koff;
            bf[0][0] = w0[(size_t)c * 9]; bf[0][1] = w0[(size_t)(c + 1) * 9];
            bf[1][0] = w1[(size_t)c * 9]; bf[1][1] = w1[(size_t)(c + 1) * 9];
        };
        auto loadAc = [&](v2f af[MT], int c0) {
            const int c = c0 + koff;
#pragma unroll
            for (int q = 0; q < MT; ++q) {
                v2f v;
                v[0] = ap[q][(size_t)c       * HW];
                v[1] = ap[q][(size_t)(c + 1) * HW];
                af[q] = v * avf[q];
            }
        };
        auto domm = [&](v2f af[MT], v2f bf[2]) {
#pragma unroll
            for (int q = 0; q < MT; ++q) {
                acc[q][0] = wmma_f32(af[q], bf[0], acc[q][0]);
                acc[q][1] = wmma_f32(af[q], bf[1], acc[q][1]);
            }
        };

        v2f b0[2], a0[MT], b1[2], a1[MT];
        loadBc(b0, 0); loadAc(a0, 0);
        int c0 = 0;
        for (; c0 + 8 < Cin; c0 += 8) {
            loadBc(b1, c0 + 4); loadAc(a1, c0 + 4);
            domm(a0, b0);
            loadBc(b0, c0 + 8); loadAc(a0, c0 + 8);
            domm(a1, b1);
        }
        loadBc(b1, c0 + 4); loadAc(a1, c0 + 4);
        domm(a0, b0);
        domm(a1, b1);
    }

#pragma unroll
    for (int q = 0; q < MT; ++q)
#pragma unroll
        for (int nf = 0; nf < 2; ++nf)
#pragma unroll
            for (int r = 0; r < 8; ++r) {
                const int m = m_base + q * 16 + r + half * 8;
                const int n = n0 + nf * 16 + row16;
                if (m < M) {
                    float v = acc[q][nf][r] + bias[n];
                    if (relu) v = fmaxf(v, 0.f);
                    out[(size_t)m * N + n] = v;
                }
            }
}

// ---------------------------------------------------------------------------
// RPN heads: 1x1 convs (18 cls + 36 bbox channels) over f stored (pos, 512).
// ---------------------------------------------------------------------------
__global__ void heads1x1(const float* __restrict__ f,
                         const float* __restrict__ cls_w, const float* __restrict__ cls_b,
                         const float* __restrict__ bbox_w, const float* __restrict__ bbox_b,
                         float* __restrict__ head) {
    const int ch = blockIdx.x;
    const int p  = threadIdx.x;          // 256 positions
    const float* w; float b;
    if (ch < 18) { w = cls_w  + (size_t)ch * 512;        b = cls_b[ch]; }
    else         { w = bbox_w + (size_t)(ch - 18) * 512; b = bbox_b[ch - 18]; }
    float s = b;
    const float* fp = f + (size_t)p * 512;
    for (int c = 0; c < 512; ++c) s += fp[c] * w[c];
    head[(size_t)ch * 256 + p] = s;
}

// ---------------------------------------------------------------------------
// Anchor generation + objectness softmax + box decode. 2304 proposals.
// ---------------------------------------------------------------------------
__global__ void proposals_kernel(const float* __restrict__ head,
                                 float* __restrict__ boxes, float* __restrict__ scores) {
    const int t = blockIdx.x * blockDim.x + threadIdx.x;
    if (t >= 2304) return;
    const int a = t >> 8;
    const int p = t & 255;
    const int y = p >> 4, x = p & 15;

    const float A0 = head[(size_t)a * 256 + p];
    const float A1 = head[(size_t)(9 + a) * 256 + p];
    scores[t] = 1.f / (1.f + expf(A0 - A1));

    const float d0 = head[(size_t)(18 + a * 4 + 0) * 256 + p];
    const float d1 = head[(size_t)(18 + a * 4 + 1) * 256 + p];
    const float d2 = head[(size_t)(18 + a * 4 + 2) * 256 + p];
    const float d3 = head[(size_t)(18 + a * 4 + 3) * 256 + p];

    const float scl[3] = {8.f, 16.f, 32.f};
    const float sq[3]  = {0.70710678f, 1.f, 1.41421356f};
    const int s = a / 3, r = a - s * 3;
    const float wa = scl[s] * sq[r];
    const float ha = scl[s] / sq[r];
    const float cx = x * 32.f, cy = y * 32.f;
    const float px = cx + d0 * wa;
    const float py = cy + d1 * ha;
    const float pw = expf(d2) * wa;
    const float ph = expf(d3) * ha;
    boxes[t * 4 + 0] = px - 0.5f * pw;
    boxes[t * 4 + 1] = py - 0.5f * ph;
    boxes[t * 4 + 2] = px + 0.5f * pw;
    boxes[t * 4 + 3] = py + 0.5f * ph;
}

// ---------------------------------------------------------------------------
// Single-workgroup bitonic sort (desc) + sequential greedy NMS + top-100.
// ---------------------------------------------------------------------------
__global__ __launch_bounds__(1024)
void nms_topk(const float* __restrict__ boxes, const float* __restrict__ scores,
              float* __restrict__ fb) {
    __shared__ float ss[4096];
    __shared__ int   si[4096];
    __shared__ int   keep[2304];
    const int tid = threadIdx.x;

    for (int i = tid; i < 4096; i += 1024) { ss[i] = (i < 2304) ? scores[i] : -1e30f; si[i] = i; }
    for (int i = tid; i < 2304; i += 1024) keep[i] = 1;
    __syncthreads();

    for (int k = 2; k <= 4096; k <<= 1) {
        for (int j = k >> 1; j > 0; j >>= 1) {
            for (int i = tid; i < 4096; i += 1024) {
                const int ixj = i ^ j;
                if (ixj > i) {
                    const bool desc = ((i & k) == 0);
                    const float a = ss[i], b = ss[ixj];
                    const bool doswap = desc ? (a < b) : (a > b);
                    if (doswap) {
                        ss[i] = b; ss[ixj] = a;
                        const int ti = si[i]; si[i] = si[ixj]; si[ixj] = ti;
                    }
                }
            }
            __syncthreads();
        }
    }

    for (int i = 0; i < 2304; ++i) {
        __syncthreads();
        if (!keep[i]) continue;
        const int bi = si[i];
        const float x1i = boxes[bi * 4 + 0], y1i = boxes[bi * 4 + 1];
        const float x2i = boxes[bi * 4 + 2], y2i = boxes[bi * 4 + 3];
        const float areai = (x2i - x1i) * (y2i - y1i);
        for (int j = i + 1 + tid; j < 2304; j += 1024) {
            if (!keep[j]) continue;
            const int bj = si[j];
            const float bx1 = boxes[bj * 4 + 0], by1 = boxes[bj * 4 + 1];
            const float bx2 = boxes[bj * 4 + 2], by2 = boxes[bj * 4 + 3];
            const float xx1 = fmaxf(x1i, bx1), yy1 = fmaxf(y1i, by1);
            const float xx2 = fminf(x2i, bx2), yy2 = fminf(y2i, by2);
            const float inter = fmaxf(xx2 - xx1, 0.f) * fmaxf(yy2 - yy1, 0.f);
            const float areaj = (bx2 - bx1) * (by2 - by1);
            const float iou = inter / (areai + areaj - inter + 1e-9f);
            if (iou > 0.5f) keep[j] = 0;
        }
    }
    __syncthreads();

    if (tid == 0) {
        int cnt = 0;
        for (int i = 0; i < 2304 && cnt < 100; ++i)
            if (keep[i]) {
                const int b = si[i];
                fb[cnt * 4 + 0] = boxes[b * 4 + 0];
                fb[cnt * 4 + 1] = boxes[b * 4 + 1];
                fb[cnt * 4 + 2] = boxes[b * 4 + 2];
                fb[cnt * 4 + 3] = boxes[b * 4 + 3];
                ++cnt;
            }
        for (; cnt < 100; ++cnt) {
            fb[cnt * 4 + 0] = 0.f; fb[cnt * 4 + 1] = 0.f;
            fb[cnt * 4 + 2] = 0.f; fb[cnt * 4 + 3] = 0.f;
        }
    }
}

// ---------------------------------------------------------------------------
// ROI Align: 100 rois x 49 pool cells (grid), 2048 channels (threads).
// ---------------------------------------------------------------------------
__global__ void roi_align_kernel(const float* __restrict__ x,
                                 const float* __restrict__ fb,
                                 float* __restrict__ out) {
    const int roi  = blockIdx.x;
    const int cell = blockIdx.y;
    const int py = cell / 7, px = cell % 7;
    const float r0 = fb[roi * 4 + 0] * 0.0625f;
    const float r1 = fb[roi * 4 + 1] * 0.0625f;
    const float r2 = fb[roi * 4 + 2] * 0.0625f;
    const float r3 = fb[roi * 4 + 3] * 0.0625f;

    int yi0[4], yi1[4], xi0[4], xi1[4];
    float ly[4], lx[4];
#pragma unroll
    for (int s = 0; s < 4; ++s) {
        const int sy = py * 2 + (s >> 1);
        const int sx = px * 2 + (s & 1);
        const float ty = (sy + 0.5f) / 14.f;
        const float tx = (sx + 0.5f) / 14.f;
        float gy = r1 + ty * (r3 - r1);
        float gx = r0 + tx * (r2 - r0);
        gy = fminf(fmaxf(gy, 0.f), 15.f);
        gx = fminf(fmaxf(gx, 0.f), 15.f);
        const float fy = floorf(gy), fx = floorf(gx);
        yi0[s] = (int)fy; xi0[s] = (int)fx;
        yi1[s] = yi0[s] + 1 > 15 ? 15 : yi0[s] + 1;
        xi1[s] = xi0[s] + 1 > 15 ? 15 : xi0[s] + 1;
        ly[s] = gy - fy; lx[s] = gx - fx;
    }

    for (int c = threadIdx.x; c < 2048; c += blockDim.x) {
        const float* fc = x + (size_t)c * 256;
        float acc = 0.f;
#pragma unroll
        for (int s = 0; s < 4; ++s) {
            const float v00 = fc[yi0[s] * 16 + xi0[s]];
            const float v01 = fc[yi0[s] * 16 + xi1[s]];
            const float v10 = fc[yi1[s] * 16 + xi0[s]];
            const float v11 = fc[yi1[s] * 16 + xi1[s]];
            acc += (1.f - ly[s]) * (1.f - lx[s]) * v00
                 + (1.f - ly[s]) * lx[s]         * v01
                 + ly[s]         * (1.f - lx[s]) * v10
                 + ly[s]         * lx[s]         * v11;
        }
        out[((size_t)roi * 2048 + c) * 49 + cell] = acc * 0.25f;
    }
}

// ---------------------------------------------------------------------------
// Small dense layer: out(MxN) = A(MxK) @ W(KxN) + bias
// ---------------------------------------------------------------------------
__global__ void fc_small(const float* __restrict__ A, const float* __restrict__ W,
                         const float* __restrict__ bias, float* __restrict__ out,
                         int M, int N, int K) {
    const int id = blockIdx.x * blockDim.x + threadIdx.x;
    if (id >= M * N) return;
    const int m = id / N, n = id - m * N;
    float s = bias[n];
    const float* a = A + (size_t)m * K;
    for (int k = 0; k < K; ++k) s += a[k] * W[(size_t)k * N + n];
    out[id] = s;
}

// ---------------------------------------------------------------------------
// Mask head tail: relu(up + mt_b) then 1x1 conv (256 -> 1).
// ---------------------------------------------------------------------------
__global__ void mask_final(const float* __restrict__ up,
                           const float* __restrict__ mt_b,
                           const float* __restrict__ m2_w, const float* __restrict__ m2_b,
                           float* __restrict__ out) {
    const int id = blockIdx.x * blockDim.x + threadIdx.x;
    if (id >= 100 * 196) return;
    const int roi = id / 196;
    const int q   = id - roi * 196;
    const int y2 = q / 14, x2 = q - (q / 14) * 14;
    const int h = y2 >> 1, i = y2 & 1;
    const int w = x2 >> 1, j = x2 & 1;
    const size_t base = ((size_t)roi * 49 + h * 7 + w) * 1024 + (i * 2 + j);
    float acc = m2_b[0];
    for (int o = 0; o < 256; ++o) {
        float v = up[base + (size_t)o * 4] + mt_b[o];
        v = fmaxf(v, 0.f);
        acc += m2_w[o] * v;
    }
    out[(size_t)roi * 196 + y2 * 14 + x2] = acc;
}

// ---------------------------------------------------------------------------
// Launcher
// ---------------------------------------------------------------------------
extern "C" void kernel_launch(void* const* d_in, const int* in_sizes, int n_in,
                              void* d_out, int out_size, void* d_ws, size_t ws_size,
                              hipStream_t stream) {
    const float* x      = (const float*)d_in[0];
    const float* rpn_w  = (const float*)d_in[1];
    const float* rpn_b  = (const float*)d_in[2];
    const float* cls_w  = (const float*)d_in[3];
    const float* cls_b  = (const float*)d_in[4];
    const float* bbox_w = (const float*)d_in[5];
    const float* bbox_b = (const float*)d_in[6];
    const float* fc1c_w = (const float*)d_in[7];
    const float* fc1c_b = (const float*)d_in[8];
    const float* fc2c_w = (const float*)d_in[9];
    const float* fc2c_b = (const float*)d_in[10];
    const float* fc1b_w = (const float*)d_in[11];
    const float* fc1b_b = (const float*)d_in[12];
    const float* fc2b_w = (const float*)d_in[13];
    const float* fc2b_b = (const float*)d_in[14];
    const float* m1_w   = (const float*)d_in[15];
    const float* m1_b   = (const float*)d_in[16];
    const float* mt_w   = (const float*)d_in[17];
    const float* mt_b   = (const float*)d_in[18];
    const float* m2_w   = (const float*)d_in[19];
    const float* m2_b   = (const float*)d_in[20];
    float* out = (float*)d_out;

    float* ws = (float*)d_ws;
    size_t off = 0;
    float* f       = ws + off; off += (size_t)256 * 512;
    float* head    = ws + off; off += (size_t)54 * 256;
    float* boxes   = ws + off; off += (size_t)2304 * 4;
    float* scores  = ws + off; off += (size_t)2304;
    float* fbuf    = ws + off; off += (size_t)112 * 4;
    float* roifeat = ws + off; off += (size_t)100 * 2048 * 49;   // == flat (100 x 100352)
    float* hidc    = ws + off; off += (size_t)100 * 4096;
    float* hidb    = ws + off; off += (size_t)100 * 4096;
    float* mrelu   = ws + off; off += (size_t)4900 * 256;
    float* up      = ws + off; off += (size_t)4900 * 1024;
    float* pbuf    = ws + off; off += (size_t)8 * 100 * 4096;    // K-split partials
    (void)ws_size; (void)in_sizes; (void)n_in; (void)out_size;

    const int KS = 8;                  // K-split for FC1 (100352/8 = 12544, %8==0)
    const int MN = 100 * 4096;

    // 1) RPN 3x3 conv 2048->512 + relu   (M=256, K=18432, N=512)
    conv3x3_gemm_wmma<2><<<dim3(8, 16), 32, 0, stream>>>(
        x, 1, 2048, 16, 16, rpn_w, rpn_b, f, 512, 1);

    // 2) 1x1 heads
    heads1x1<<<54, 256, 0, stream>>>(f, cls_w, cls_b, bbox_w, bbox_b, head);

    // 3) anchors + softmax + decode
    proposals_kernel<<<(2304 + 255) / 256, 256, 0, stream>>>(head, boxes, scores);

    // 4) sort + NMS + top-100
    nms_topk<<<1, 1024, 0, stream>>>(boxes, scores, fbuf);

    // 5) ROI align -> (100, 2048, 7, 7)
    roi_align_kernel<<<dim3(100, 49), 256, 0, stream>>>(x, fbuf, roifeat);

    // 6) FC1 cls: K-split x8 (1024 waves streaming the 1.64GB weight matrix once)
    gemm_wmma<7><<<dim3(1, 128, KS), 32, 0, stream>>>(
        roifeat, 100352, fc1c_w, 4096, 1, nullptr, pbuf, 100, 4096, 100352, 0, 100352 / KS);
    reduce_partials<<<(MN + 255) / 256, 256, 0, stream>>>(pbuf, KS, fc1c_b, hidc, MN, 4096, 1);

    // 7) FC2 cls -> out[0 .. 9100)
    fc_small<<<(100 * 91 + 255) / 256, 256, 0, stream>>>(
        hidc, fc2c_w, fc2c_b, out, 100, 91, 4096);

    // 8) FC1 box + FC2 box -> out[9100 .. 9500)
    gemm_wmma<7><<<dim3(1, 128, KS), 32, 0, stream>>>(
        roifeat, 100352, fc1b_w, 4096, 1, nullptr, pbuf, 100, 4096, 100352, 0, 100352 / KS);
    reduce_partials<<<(MN + 255) / 256, 256, 0, stream>>>(pbuf, KS, fc1b_b, hidb, MN, 4096, 1);
    fc_small<<<(100 * 4 + 255) / 256, 256, 0, stream>>>(
        hidb, fc2b_w, fc2b_b, out + 9100, 100, 4, 4096);

    // 9) Mask 3x3 conv 2048->256 + relu   (M=4900, K=18432, N=256)
    conv3x3_gemm_wmma<4><<<dim3((4900 + 63) / 64, 8), 32, 0, stream>>>(
        roifeat, 100, 2048, 7, 7, m1_w, m1_b, mrelu, 256, 1);

    // 10) Transposed conv as GEMM: (4900 x 256) @ (256 x 1024)
    gemm_wmma<4><<<dim3((4900 + 63) / 64, 32, 1), 32, 0, stream>>>(
        mrelu, 256, mt_w, 1024, 1, nullptr, up, 4900, 1024, 256, 0, 256);

    // 11) relu(up + mt_b) then 1x1 conv -> masks, out[9500 .. 29100)
    mask_final<<<(100 * 196 + 255) / 256, 256, 0, stream>>>(
        up, mt_b, m2_w, m2_b, out + 9500);
}